// GCMCGraphConv_51049981280527
// MI455X (gfx1250) — compile-verified
//
#include <hip/hip_runtime.h>

// GCMC graph conv for MI455X (gfx1250, wave32):
//   out[dst, :] += weight[src, :] * cj[src] * ci[dst]   (fused normalize)
// N=100000, F=128, E=1600000. Memory/atomic-bound scatter-gather.
//
// Batched design: each wave32 owns 32 edges.
//   - src/dst for the batch: two coalesced 128B loads (lane l = edge base+l)
//   - cj[src]*ci[dst]: two 4B gathers + one VALU mul -> per-lane scale
//   - weight rows for all 32 edges warmed with per-lane global_prefetch_b8
//   - unrolled per-edge body: v_readlane (imm lane) -> global_load_b128
//     row gather -> 4x no-return global_atomic_add_f32 into out[dst,:]
// 32 lanes x float4 = exactly one 128-float row per edge, fully coalesced.

#define FEAT 128
#define EPW 32                       // edges per wave
#define WAVES_PER_BLOCK 8
#define BLOCK_THREADS (WAVES_PER_BLOCK * 32)

__global__ void gcmc_zero_kernel(float* __restrict__ out, long n4) {
    long i = (long)blockIdx.x * blockDim.x + threadIdx.x;
    long stride = (long)gridDim.x * blockDim.x;
    float4 z = {0.0f, 0.0f, 0.0f, 0.0f};
    for (; i < n4; i += stride) {
        ((float4*)out)[i] = z;
    }
}

__device__ __forceinline__ void gcmc_edge_body(
    const float* __restrict__ weight, float* __restrict__ out,
    int s, int d, float sc, int lane) {
    const float4 w = ((const float4*)(weight + (long)s * FEAT))[lane];
    float* __restrict__ orow = out + (long)d * FEAT + 4 * lane;
    atomicAdd(orow + 0, w.x * sc);
    atomicAdd(orow + 1, w.y * sc);
    atomicAdd(orow + 2, w.z * sc);
    atomicAdd(orow + 3, w.w * sc);
}

__global__ void __launch_bounds__(BLOCK_THREADS)
gcmc_edge_scatter_kernel(const float* __restrict__ weight,
                         const float* __restrict__ cj,
                         const float* __restrict__ ci,
                         const int*   __restrict__ src,
                         const int*   __restrict__ dst,
                         float*       __restrict__ out,
                         int E) {
    const int lane  = threadIdx.x & 31;
    const int gwave = blockIdx.x * WAVES_PER_BLOCK + (threadIdx.x >> 5);
    const long base = (long)gwave * EPW;
    if (base >= E) return;

    // Coalesced batch index loads: lane l owns edge base+l.
    const int  idx   = (int)base + lane;
    const bool valid = idx < E;
    const int  vs = valid ? src[idx] : 0;
    const int  vd = valid ? dst[idx] : 0;

    // Warm all 32 scattered weight rows for this batch while the cj/ci
    // gathers are in flight. Per-lane addresses -> one global_prefetch_b8
    // instruction covers 32 rows; 4 issues cover the full 512B row span.
    const float* __restrict__ wrow0 = weight + (long)vs * FEAT;
    __builtin_prefetch(wrow0 +  0, 0, 3);
    __builtin_prefetch(wrow0 + 32, 0, 3);
    __builtin_prefetch(wrow0 + 64, 0, 3);
    __builtin_prefetch(wrow0 + 96, 0, 3);

    // Fused normalization: (sum msgs)*ci[d] == sum(msgs*ci[d]).
    const float scj = valid ? cj[vs] : 0.0f;   // 4B gather
    const float sci = valid ? ci[vd] : 0.0f;   // 4B gather
    const int iscale = __builtin_bit_cast(int, scj * sci);

    if (base + EPW <= (long)E) {
        // Fast path: full batch, no per-edge guards. Fully unrolled so the
        // v_readlane lane index is an immediate and the scheduler can
        // pipeline the 32 independent row-loads against the atomics.
#pragma unroll
        for (int i = 0; i < EPW; ++i) {
            const int   s  = __builtin_amdgcn_readlane(vs, i);
            const int   d  = __builtin_amdgcn_readlane(vd, i);
            const float sc = __builtin_bit_cast(
                float, __builtin_amdgcn_readlane(iscale, i));
            gcmc_edge_body(weight, out, s, d, sc, lane);
        }
    } else {
        const int rem = (int)(E - base);
#pragma unroll
        for (int i = 0; i < EPW; ++i) {
            if (i < rem) {
                const int   s  = __builtin_amdgcn_readlane(vs, i);
                const int   d  = __builtin_amdgcn_readlane(vd, i);
                const float sc = __builtin_bit_cast(
                    float, __builtin_amdgcn_readlane(iscale, i));
                gcmc_edge_body(weight, out, s, d, sc, lane);
            }
        }
    }
}

extern "C" void kernel_launch(void* const* d_in, const int* in_sizes, int n_in,
                              void* d_out, int out_size, void* d_ws, size_t ws_size,
                              hipStream_t stream) {
    const float* weight = (const float*)d_in[0];   // [N, 128]
    const float* cj     = (const float*)d_in[1];   // [N, 1]
    const float* ci     = (const float*)d_in[2];   // [N, 1]
    const int*   src    = (const int*)d_in[3];     // [E]
    const int*   dst    = (const int*)d_in[4];     // [E]
    float* out = (float*)d_out;                    // [N, 128]

    const int E = in_sizes[3];

    // 1) Zero the accumulator (d_out is poisoned by the harness).
    const long n4 = (long)out_size / 4;            // float4 count
    int zblocks = (int)((n4 + 255) / 256);
    if (zblocks > 65535) zblocks = 65535;          // grid-stride covers rest
    gcmc_zero_kernel<<<zblocks, 256, 0, stream>>>(out, n4);

    // 2) Fused gather + scale + scatter-add; 32 edges per wave, 8 waves/block.
    const long edges_per_block = (long)WAVES_PER_BLOCK * EPW;
    const int eblocks = (int)((E + edges_per_block - 1) / edges_per_block);
    gcmc_edge_scatter_kernel<<<eblocks, BLOCK_THREADS, 0, stream>>>(
        weight, cj, ci, src, dst, out, E);
}